// GAT_46961172414533
// MI455X (gfx1250) — compile-verified
//
#include <hip/hip_runtime.h>
#include <hip/hip_bf16.h>

#define N_NODES 50000
#define N_EDGES 800000
#define D 128

typedef __attribute__((ext_vector_type(2))) float v2f;
typedef __attribute__((ext_vector_type(8))) float v8f;

// ---------------- monotone float <-> uint encoding for atomic segment-max ----
__device__ __forceinline__ unsigned fenc(float f) {
    unsigned u = __float_as_uint(f);
    return (u & 0x80000000u) ? ~u : (u | 0x80000000u);
}
__device__ __forceinline__ float fdec(unsigned u) {
    return (u & 0x80000000u) ? __uint_as_float(u & 0x7FFFFFFFu)
                             : __uint_as_float(~u);
}

// ---------------- zero fill -------------------------------------------------
__global__ void gat_zero_u32(unsigned* __restrict__ p, int n) {
    int i = blockIdx.x * blockDim.x + threadIdx.x;
    if (i < n) p[i] = 0u;
}

// ---------------- dense GEMM: Z[N,128] = X[N,128] @ W[128,128] (fp32 WMMA) --
// Block: 256 threads = 8 waves. Block handles 16 rows, wave w handles cols
// [16w,16w+16). K-loop in steps of 4 using V_WMMA_F32_16X16X4_F32.
// A (16x4 f32): lanes 0-15 -> {K0,K1} of row lane; lanes 16-31 -> {K2,K3}.
// B (4x16 f32): lanes 0-15 -> rows K0(K1) col lane; lanes 16-31 -> K2(K3).
// C (16x16 f32): VGPR r -> row r (lanes 0-15) / row r+8 (lanes 16-31).
__global__ void __launch_bounds__(256)
gat_gemm_f32(const float* __restrict__ X, const float* __restrict__ W,
             float* __restrict__ Z) {
    __shared__ float xs[16][D];            // 8 KB tile of X
    const int m0 = blockIdx.x * 16;
    for (int i = threadIdx.x; i < 16 * D; i += 256) {
        int r = i >> 7, c = i & (D - 1);
        xs[r][c] = X[(m0 + r) * D + c];
    }
    __syncthreads();

    const int lane  = threadIdx.x & 31;
    const int wave  = threadIdx.x >> 5;    // 0..7
    const int n0    = wave * 16;
    const int row   = lane & 15;
    const int khalf = (lane >> 4) * 2;     // 0 for lanes 0-15, 2 for 16-31

    v8f acc = {};
    #pragma unroll 4
    for (int k0 = 0; k0 < D; k0 += 4) {
        v2f a, b;
        a.x = xs[row][k0 + khalf + 0];
        a.y = xs[row][k0 + khalf + 1];
        b.x = W[(k0 + khalf + 0) * D + n0 + row];
        b.y = W[(k0 + khalf + 1) * D + n0 + row];
        acc = __builtin_amdgcn_wmma_f32_16x16x4_f32(
            /*neg_a=*/false, a, /*neg_b=*/false, b,
            /*c_mod=*/(short)0, acc, /*reuse_a=*/false, /*reuse_b=*/false);
    }

    const int mbase = m0 + (lane >> 4) * 8;
    const int col   = n0 + row;
    #pragma unroll
    for (int r = 0; r < 8; ++r)
        Z[(mbase + r) * D + col] = acc[r];
}

// ---------------- attention scores: s_src = z.a_lo, s_dst = z.a_hi ----------
__global__ void __launch_bounds__(256)
gat_sdots(const float* __restrict__ Z, const float* __restrict__ a,
          float* __restrict__ s_src, float* __restrict__ s_dst, int n) {
    int node = blockIdx.x * (blockDim.x >> 5) + (threadIdx.x >> 5);
    int lane = threadIdx.x & 31;
    if (node >= n) return;
    const float* z = Z + node * D;
    float a0 = 0.f, a1 = 0.f;
    #pragma unroll
    for (int j = lane; j < D; j += 32) {
        float zv = z[j];
        a0 += zv * a[j];
        a1 += zv * a[D + j];
    }
    #pragma unroll
    for (int off = 16; off > 0; off >>= 1) {
        a0 += __shfl_down(a0, off, 32);
        a1 += __shfl_down(a1, off, 32);
    }
    if (lane == 0) { s_src[node] = a0; s_dst[node] = a1; }
}

// ---------------- edge pass 1: leaky_relu score + segment max ---------------
__global__ void __launch_bounds__(256)
gat_edge_max(const float* __restrict__ s_src, const float* __restrict__ s_dst,
             const int* __restrict__ src, const int* __restrict__ dst,
             float* __restrict__ ebuf, unsigned* __restrict__ mbits, int ne) {
    int e = blockIdx.x * blockDim.x + threadIdx.x;
    if (e >= ne) return;
    float v = s_src[src[e]] + s_dst[dst[e]];
    v = v > 0.f ? v : 0.01f * v;           // leaky_relu slope 0.01
    ebuf[e] = v;
    atomicMax(&mbits[dst[e]], fenc(v));
}

// ---------------- edge pass 2: exp(e - m[dst]) + segment sum ----------------
__global__ void __launch_bounds__(256)
gat_edge_expsum(const int* __restrict__ dst, const unsigned* __restrict__ mbits,
                float* __restrict__ ebuf, float* __restrict__ denom, int ne) {
    int e = blockIdx.x * blockDim.x + threadIdx.x;
    if (e >= ne) return;
    int d = dst[e];
    float ex = expf(ebuf[e] - fdec(mbits[d]));
    ebuf[e] = ex;
    atomicAdd(&denom[d], ex);
}

// ---------------- scatter: agg[dst] += (ex/denom[dst]) * z[src] -------------
// One wave per edge, float4 gather, 4 fp32 atomics per lane (L2-resident).
__global__ void __launch_bounds__(256)
gat_scatter(const float* __restrict__ Z, const float* __restrict__ ebuf,
            const float* __restrict__ denom, const int* __restrict__ src,
            const int* __restrict__ dst, float* __restrict__ agg, int ne) {
    int e    = blockIdx.x * (blockDim.x >> 5) + (threadIdx.x >> 5);
    int lane = threadIdx.x & 31;
    if (e >= ne) return;
    int s = src[e], d = dst[e];
    float alpha = ebuf[e] / denom[d];
    float4 v = ((const float4*)(Z + s * D))[lane];
    float* o = agg + d * D + lane * 4;
    atomicAdd(o + 0, alpha * v.x);
    atomicAdd(o + 1, alpha * v.y);
    atomicAdd(o + 2, alpha * v.z);
    atomicAdd(o + 3, alpha * v.w);
}

// ---------------- elementwise ELU ------------------------------------------
__global__ void __launch_bounds__(256)
gat_elu(const float* __restrict__ in, float* __restrict__ out, int n) {
    int i = blockIdx.x * blockDim.x + threadIdx.x;
    if (i < n) {
        float v = in[i];
        out[i] = v > 0.f ? v : expm1f(v);
    }
}

// ---------------- column sums for mean readout ------------------------------
__global__ void __launch_bounds__(128)
gat_colsum(const float* __restrict__ X, float* __restrict__ hg, int n) {
    int f = threadIdx.x;                   // feature 0..127 (coalesced)
    float acc = 0.f;
    for (int i = blockIdx.x; i < n; i += gridDim.x)
        acc += X[i * D + f];
    atomicAdd(&hg[f], acc);
}

// ---------------- tiny MLP readout: 128 -> 64 -> 32 -> 3 --------------------
__global__ void __launch_bounds__(128)
gat_mlp(const float* __restrict__ hg,
        const float* __restrict__ M0w, const float* __restrict__ M0b,
        const float* __restrict__ M1w, const float* __restrict__ M1b,
        const float* __restrict__ M2w, const float* __restrict__ M2b,
        float* __restrict__ out, float invN) {
    __shared__ float hgm[128];
    __shared__ float y0[64];
    __shared__ float y1[32];
    int t = threadIdx.x;
    hgm[t] = hg[t] * invN;
    __syncthreads();
    if (t < 64) {
        float acc = M0b[t];
        for (int k = 0; k < 128; ++k) acc += hgm[k] * M0w[k * 64 + t];
        y0[t] = acc > 0.f ? acc : 0.f;
    }
    __syncthreads();
    if (t < 32) {
        float acc = M1b[t];
        for (int k = 0; k < 64; ++k) acc += y0[k] * M1w[k * 32 + t];
        y1[t] = acc > 0.f ? acc : 0.f;
    }
    __syncthreads();
    if (t < 3) {
        float acc = M2b[t];
        for (int k = 0; k < 32; ++k) acc += y1[k] * M2w[k * 3 + t];
        out[t] = acc;
    }
}

extern "C" void kernel_launch(void* const* d_in, const int* in_sizes, int n_in,
                              void* d_out, int out_size, void* d_ws, size_t ws_size,
                              hipStream_t stream) {
    (void)in_sizes; (void)n_in; (void)out_size; (void)ws_size;
    const float* h   = (const float*)d_in[0];
    const int*   src = (const int*)d_in[1];
    const int*   dst = (const int*)d_in[2];
    const float* Wl[3] = {(const float*)d_in[3], (const float*)d_in[5], (const float*)d_in[7]};
    const float* al[3] = {(const float*)d_in[4], (const float*)d_in[6], (const float*)d_in[8]};
    const float* M0w = (const float*)d_in[9];
    const float* M0b = (const float*)d_in[10];
    const float* M1w = (const float*)d_in[11];
    const float* M1b = (const float*)d_in[12];
    const float* M2w = (const float*)d_in[13];
    const float* M2b = (const float*)d_in[14];
    float* out = (float*)d_out;

    // workspace layout (floats): z | agg | x | s_src | s_dst | mbits | denom | ebuf | hg
    float*    z     = (float*)d_ws;
    float*    agg   = z + (size_t)N_NODES * D;
    float*    x     = agg + (size_t)N_NODES * D;
    float*    ssrc  = x + (size_t)N_NODES * D;
    float*    sdst  = ssrc + N_NODES;
    unsigned* mbits = (unsigned*)(sdst + N_NODES);
    float*    denom = (float*)(mbits + N_NODES);
    float*    ebuf  = denom + N_NODES;
    float*    hg    = ebuf + N_EDGES;

    const int nodeBlocks16 = N_NODES / 16;                       // 3125 (exact)
    const int waveBlocksN  = (N_NODES + 7) / 8;                  // wave-per-node, 8 waves/block
    const int edgeBlocks   = (N_EDGES + 255) / 256;
    const int waveBlocksE  = (N_EDGES + 7) / 8;                  // wave-per-edge
    const int elemBlocks   = (N_NODES * D + 255) / 256;

    const float* xin = h;
    for (int l = 0; l < 3; ++l) {
        // z = xin @ W  (fp32 WMMA)
        gat_gemm_f32<<<nodeBlocks16, 256, 0, stream>>>(xin, Wl[l], z);
        // attention source/dest scores
        gat_sdots<<<waveBlocksN, 256, 0, stream>>>(z, al[l], ssrc, sdst, N_NODES);
        // clear segment-max bits, denom, agg
        gat_zero_u32<<<(N_NODES + 255) / 256, 256, 0, stream>>>(mbits, N_NODES);
        gat_zero_u32<<<(N_NODES + 255) / 256, 256, 0, stream>>>((unsigned*)denom, N_NODES);
        gat_zero_u32<<<elemBlocks, 256, 0, stream>>>((unsigned*)agg, N_NODES * D);
        // segment softmax over edges
        gat_edge_max<<<edgeBlocks, 256, 0, stream>>>(ssrc, sdst, src, dst, ebuf, mbits, N_EDGES);
        gat_edge_expsum<<<edgeBlocks, 256, 0, stream>>>(dst, mbits, ebuf, denom, N_EDGES);
        // weighted message scatter (L2-resident atomics)
        gat_scatter<<<waveBlocksE, 256, 0, stream>>>(z, ebuf, denom, src, dst, agg, N_EDGES);
        if (l < 2) {
            gat_elu<<<elemBlocks, 256, 0, stream>>>(agg, x, N_NODES * D);
            xin = x;
        }
    }

    // mean readout + MLP
    gat_zero_u32<<<1, 128, 0, stream>>>((unsigned*)hg, D);
    gat_colsum<<<512, 128, 0, stream>>>(agg, hg, N_NODES);
    gat_mlp<<<1, 128, 0, stream>>>(hg, M0w, M0b, M1w, M1b, M2w, M2b, out,
                                   1.0f / (float)N_NODES);
}